// LinConv_77996606095426
// MI455X (gfx1250) — compile-verified
//
#include <hip/hip_runtime.h>

#define N_NODES   100000
#define N_EDGES   3200000
#define SEQ_LEN   20
#define D         128
#define NUM_LAYERS 3
#define SCALE     0.25f   // 1/(NUM_LAYERS+1)

typedef __attribute__((ext_vector_type(2))) float v2f;
typedef __attribute__((ext_vector_type(8))) float v8f;

// Phase 1: features[n,d] = (1/len[n]) * sum_t seqEmbs[n,t,d], via V_WMMA_F32_16X16X4_F32
// with A = ones(16x4). Every output row of D is the column-sum of B, so row M=0
// (C VGPR0, lanes 0-15) gives sum over 4 timesteps of 16 consecutive d's.
// One wave per node: 8 d-chunks x 5 t-chunks (SEQ_LEN = 5*4 exactly).
// Also zeroes the first scatter target (zbuf) and initializes out = SCALE*features.
__global__ void __launch_bounds__(256) mean_wmma_kernel(
    const float* __restrict__ seqEmbs, const float* __restrict__ seqLens,
    float* __restrict__ feat, float* __restrict__ zbuf, float* __restrict__ out)
{
    const int wave = blockIdx.x * (blockDim.x >> 5) + (threadIdx.x >> 5);
    const int lane = threadIdx.x & 31;
    const int n = wave;
    if (n >= N_NODES) return;   // wave-uniform (grid divides exactly; EXEC stays all-1s)

    const float invLen = 1.0f / seqLens[n];
    const float* __restrict__ E = seqEmbs + (size_t)n * (SEQ_LEN * D);
    const int tOff  = (lane >> 4) * 2;   // lanes 0-15 hold K=0,1 ; lanes 16-31 hold K=2,3
    const int dLane = lane & 15;         // column N within the 16-wide d-chunk

    v2f a; a.x = 1.0f; a.y = 1.0f;       // A = ones (16x4 striped over lanes)

    float* __restrict__ frow = feat + (size_t)n * D;
    float* __restrict__ orow = out  + (size_t)n * D;

    // zero this node's row in the first scatter target (32 lanes x float4 = 128)
    ((float4*)(zbuf + (size_t)n * D))[lane] = make_float4(0.f, 0.f, 0.f, 0.f);

    for (int dc = 0; dc < D / 16; ++dc) {
        const int dBase = dc * 16;
        v8f c = {};
#pragma unroll
        for (int tc = 0; tc < SEQ_LEN / 4; ++tc) {
            const int t0 = tc * 4 + tOff;
            v2f b;
            b.x = E[(size_t)t0       * D + dBase + dLane];
            b.y = E[(size_t)(t0 + 1) * D + dBase + dLane];
            // 8 args: (neg_a, A, neg_b, B, c_mod, C, reuse_a, reuse_b)
            c = __builtin_amdgcn_wmma_f32_16x16x4_f32(
                    false, a, false, b, (short)0, c, false, false);
        }
        if (lane < 16) {                 // row M=0 lives in C VGPR0, lanes 0-15
            float s = c[0] * invLen;
            frow[dBase + lane] = s;
            orow[dBase + lane] = s * SCALE;
        }
    }
}

// Phase 2 SpMM: one wave per edge. Wave gathers the full 512B source row
// (float4/lane, coalesced, L2-resident: 51.2MB << 192MB L2) and scatter-adds
// w * row into the destination row with global f32 atomics (resolved in L2).
__global__ void __launch_bounds__(256) scatter_kernel(
    const int* __restrict__ edge_index, const float* __restrict__ edge_weight,
    const float* __restrict__ src_feat, float* __restrict__ dst_feat)
{
    const long long tid = (long long)blockIdx.x * blockDim.x + threadIdx.x;
    const int e    = (int)(tid >> 5);
    const int lane = (int)(tid & 31);
    if (e >= N_EDGES) return;

    const int   s = edge_index[e];
    const int   t = edge_index[N_EDGES + e];
    const float w = edge_weight[e];

    const float4 f = ((const float4*)(src_feat + (size_t)s * D))[lane];
    float* drow = dst_feat + (size_t)t * D + lane * 4;
    atomicAdd(drow + 0, w * f.x);
    atomicAdd(drow + 1, w * f.y);
    atomicAdd(drow + 2, w * f.z);
    atomicAdd(drow + 3, w * f.w);
}

// out += SCALE * acc; also zero zbuf (the next layer's scatter target) if non-null.
__global__ void __launch_bounds__(256) accum_zero_kernel(
    float* __restrict__ out, const float* __restrict__ acc, float* __restrict__ zbuf)
{
    const size_t i = (size_t)blockIdx.x * blockDim.x + threadIdx.x;  // float4 index
    float4 a = ((const float4*)acc)[i];
    float4 o = ((float4*)out)[i];
    o.x += SCALE * a.x; o.y += SCALE * a.y;
    o.z += SCALE * a.z; o.w += SCALE * a.w;
    ((float4*)out)[i] = o;
    if (zbuf) ((float4*)zbuf)[i] = make_float4(0.f, 0.f, 0.f, 0.f);
}

extern "C" void kernel_launch(void* const* d_in, const int* in_sizes, int n_in,
                              void* d_out, int out_size, void* d_ws, size_t ws_size,
                              hipStream_t stream) {
    const int*   edge_index  = (const int*)d_in[0];    // (2, N_EDGES) int32
    const float* edge_weight = (const float*)d_in[1];  // (N_EDGES,)
    const float* seqEmbs     = (const float*)d_in[2];  // (N, SEQ_LEN, D)
    const float* seqLens     = (const float*)d_in[3];  // (N, 1)
    float* out  = (float*)d_out;                       // (N, D)
    float* buf0 = (float*)d_ws;                        // features ping
    float* buf1 = buf0 + (size_t)N_NODES * D;          // features pong (102.4MB total ws)

    // features -> buf0, out = SCALE*features, buf1 = 0
    mean_wmma_kernel<<<N_NODES / 8, 256, 0, stream>>>(seqEmbs, seqLens, buf0, buf1, out);

    float* cur = buf0;
    float* nxt = buf1;
    for (int layer = 0; layer < NUM_LAYERS; ++layer) {
        // nxt is guaranteed zeroed (by mean_wmma_kernel or previous accum_zero)
        scatter_kernel<<<(N_EDGES * 32) / 256, 256, 0, stream>>>(
            edge_index, edge_weight, cur, nxt);
        float* zb = (layer + 1 < NUM_LAYERS) ? cur : nullptr;
        accum_zero_kernel<<<(N_NODES * D / 4) / 256, 256, 0, stream>>>(out, nxt, zb);
        float* tmp = cur; cur = nxt; nxt = tmp;
    }
}